// GCN_40029095199334
// MI455X (gfx1250) — compile-verified
//
#include <hip/hip_runtime.h>

typedef __attribute__((ext_vector_type(2))) float v2f;
typedef __attribute__((ext_vector_type(8))) float v8f;

#define HID 256
#define TPB 256

// ---------------- small elementwise kernels ----------------

__global__ void fill_kernel(float* __restrict__ p, float v, int n) {
  int i = blockIdx.x * blockDim.x + threadIdx.x;
  if (i < n) p[i] = v;
}

__global__ void degree_kernel(const int* __restrict__ dst,
                              float* __restrict__ deg, int E) {
  int e = blockIdx.x * blockDim.x + threadIdx.x;
  if (e < E) unsafeAtomicAdd(&deg[dst[e]], 1.0f);
}

__global__ void rsqrt_inplace_kernel(float* __restrict__ d, int n) {
  int i = blockIdx.x * blockDim.x + threadIdx.x;
  if (i < n) d[i] = 1.0f / sqrtf(d[i]);
}

__global__ void enorm_kernel(const int* __restrict__ src,
                             const int* __restrict__ dst,
                             const float* __restrict__ dinv,
                             float* __restrict__ enorm, int E) {
  int e = blockIdx.x * blockDim.x + threadIdx.x;
  if (e < E) enorm[e] = dinv[src[e]] * dinv[dst[e]];
}

// ---------------- WMMA GEMM: H[N,256] = relu?(X[N,K]) @ W[K,256] ----------
// One wave computes a 32-row x 128-col tile (2 row-tiles x 8 col-tiles) with
// V_WMMA_F32_16X16X4_F32; each B fragment feeds two WMMAs (A-reuse halves the
// B-load bandwidth vs a 16x256 strip). Block = 8 waves = 4 row-groups x
// 2 col-groups = 128 rows x 256 cols.
// A frag (16x4 f32): lanes 0-15 hold M=0..15 with (K, K+1); lanes 16-31 hold
// (K+2, K+3) -> contiguous float2 per lane.
// B frag (4x16 f32): VGPR0 = row K (lanes 0-15) / row K+2 (lanes 16-31),
// VGPR1 = row K+1 / K+3.
// C/D (16x16 f32): VGPR r = row r (lanes 0-15) / row r+8 (lanes 16-31).
// RELU applies elementwise relu to X on load (fuses the previous layer's
// activation, saving a full read+write pass over the feature map).
template <int K, bool RELU>
__global__ __launch_bounds__(TPB) void gemm_wmma_kernel(
    const float* __restrict__ X, const float* __restrict__ W,
    float* __restrict__ H, int N) {
  const int lane = threadIdx.x & 31;
  const int wave = threadIdx.x >> 5;  // 0..7
  const int rgrp = wave >> 1;         // 0..3 (32 rows each)
  const int cgrp = wave & 1;          // 0..1 (128 cols each)
  const int row_base = blockIdx.x * 128 + rgrp * 32;
  if (row_base >= N) return;  // wave-uniform (N % 32 == 0): EXEC all-ones

  const int col_base = cgrp * 128;
  const int col = lane & 15;
  const int khalf = (lane >> 4) << 1;  // 0 for lanes 0-15, 2 for lanes 16-31
  const size_t arow0 = (size_t)(row_base + col);
  const size_t arow1 = (size_t)(row_base + 16 + col);

  v8f acc[2][8];
#pragma unroll
  for (int m = 0; m < 2; ++m)
#pragma unroll
    for (int j = 0; j < 8; ++j)
      acc[m][j] = (v8f){0.f, 0.f, 0.f, 0.f, 0.f, 0.f, 0.f, 0.f};

  for (int k = 0; k < K; k += 4) {
    v2f a0 = *(const v2f*)(X + arow0 * K + (k + khalf));
    v2f a1 = *(const v2f*)(X + arow1 * K + (k + khalf));
    if (RELU) {
      a0.x = fmaxf(a0.x, 0.f);
      a0.y = fmaxf(a0.y, 0.f);
      a1.x = fmaxf(a1.x, 0.f);
      a1.y = fmaxf(a1.y, 0.f);
    }
#pragma unroll
    for (int j = 0; j < 8; ++j) {
      v2f b;
      b.x = W[(size_t)(k + khalf) * HID + col_base + j * 16 + col];
      b.y = W[(size_t)(k + khalf + 1) * HID + col_base + j * 16 + col];
      acc[0][j] = __builtin_amdgcn_wmma_f32_16x16x4_f32(
          false, a0, false, b, (short)0, acc[0][j], false, false);
      acc[1][j] = __builtin_amdgcn_wmma_f32_16x16x4_f32(
          false, a1, false, b, (short)0, acc[1][j], false, false);
    }
  }

#pragma unroll
  for (int m = 0; m < 2; ++m) {
    const int rlo = row_base + m * 16 + (lane >> 4) * 8;
#pragma unroll
    for (int j = 0; j < 8; ++j) {
#pragma unroll
      for (int r = 0; r < 8; ++r) {
        H[(size_t)(rlo + r) * HID + col_base + j * 16 + col] = acc[m][j][r];
      }
    }
  }
}

// ---------------- aggregation ----------------

// agg[i,:] = h[i,:] * dinv[i]^2 + bias   (self-loop term + bias, pre-scatter)
__global__ void init_agg_kernel(const float* __restrict__ h,
                                const float* __restrict__ dinv,
                                const float* __restrict__ bias,
                                float* __restrict__ agg, int N) {
  int t = blockIdx.x * blockDim.x + threadIdx.x;
  int node = t >> 6;
  if (node >= N) return;
  int c = (t & 63) << 2;
  float di = dinv[node];
  float sn = di * di;
  float4 hv = *(const float4*)(h + (size_t)node * HID + c);
  float4 bv = *(const float4*)(bias + c);
  float4 o;
  o.x = fmaf(hv.x, sn, bv.x);
  o.y = fmaf(hv.y, sn, bv.y);
  o.z = fmaf(hv.z, sn, bv.z);
  o.w = fmaf(hv.w, sn, bv.w);
  *(float4*)(agg + (size_t)node * HID + c) = o;
}

// agg[dst[e],:] += h[src[e],:] * enorm[e]
// One wave per edge: edge id forced into an SGPR so src/dst/enorm become
// wave-uniform scalar loads; each lane moves 8 contiguous floats (2 x b128
// gathers + 8 hardware global_atomic_add_f32).
__global__ __launch_bounds__(TPB) void scatter_kernel(
    const float* __restrict__ h, const int* __restrict__ src,
    const int* __restrict__ dst, const float* __restrict__ enorm,
    float* __restrict__ agg, int E) {
  int e = __builtin_amdgcn_readfirstlane(blockIdx.x * (TPB / 32) +
                                         (threadIdx.x >> 5));
  if (e >= E) return;  // wave-uniform branch
  const int s = src[e];
  const int d = dst[e];
  const float w = enorm[e];
  const int c = (threadIdx.x & 31) * 8;
  const float* hp = h + (size_t)s * HID + c;
  float* ap = agg + (size_t)d * HID + c;
  float4 v0 = *(const float4*)(hp);
  float4 v1 = *(const float4*)(hp + 4);
  unsafeAtomicAdd(ap + 0, v0.x * w);
  unsafeAtomicAdd(ap + 1, v0.y * w);
  unsafeAtomicAdd(ap + 2, v0.z * w);
  unsafeAtomicAdd(ap + 3, v0.w * w);
  unsafeAtomicAdd(ap + 4, v1.x * w);
  unsafeAtomicAdd(ap + 5, v1.y * w);
  unsafeAtomicAdd(ap + 6, v1.z * w);
  unsafeAtomicAdd(ap + 7, v1.w * w);
}

// ---------------- driver ----------------

static inline int cdiv(int a, int b) { return (a + b - 1) / b; }

extern "C" void kernel_launch(void* const* d_in, const int* in_sizes, int n_in,
                              void* d_out, int out_size, void* d_ws,
                              size_t ws_size, hipStream_t stream) {
  const float* x = (const float*)d_in[0];
  const int* ei = (const int*)d_in[1];
  const float* W1 = (const float*)d_in[2];
  const float* b1 = (const float*)d_in[3];
  const float* W2 = (const float*)d_in[4];
  const float* b2 = (const float*)d_in[5];
  const float* W3 = (const float*)d_in[6];
  const float* b3 = (const float*)d_in[7];
  float* out = (float*)d_out;

  const int F = 128;
  const int N = in_sizes[0] / F;
  const int E = in_sizes[1] / 2;
  const int* src = ei;
  const int* dst = ei + E;

  // workspace layout (floats), 256-float aligned chunks
  float* wsf = (float*)d_ws;
  size_t off = 0;
  float* dinv = wsf + off;
  off += (size_t)((N + 255) / 256) * 256;
  float* enorm = wsf + off;
  off += (size_t)((E + 255) / 256) * 256;
  float* hbuf = wsf + off;
  off += (size_t)N * HID;
  float* agg = wsf + off;

  const int nodeGrid = cdiv(N, TPB);
  const int edgeGrid = cdiv(E, TPB);
  const int featGrid = cdiv(N * 64, TPB);  // node x float4 threads
  const int scatGrid = cdiv(E, TPB / 32);  // one wave per edge
  const int gemmGrid = cdiv(N, 128);       // 128 rows per block

  // degree (with self loop) -> dinv -> per-edge norm (reused by all 3 layers)
  fill_kernel<<<nodeGrid, TPB, 0, stream>>>(dinv, 1.0f, N);
  degree_kernel<<<edgeGrid, TPB, 0, stream>>>(dst, dinv, E);
  rsqrt_inplace_kernel<<<nodeGrid, TPB, 0, stream>>>(dinv, N);
  enorm_kernel<<<edgeGrid, TPB, 0, stream>>>(src, dst, dinv, enorm, E);

  // layer 1: K=128, no input relu
  gemm_wmma_kernel<128, false><<<gemmGrid, TPB, 0, stream>>>(x, W1, hbuf, N);
  init_agg_kernel<<<featGrid, TPB, 0, stream>>>(hbuf, dinv, b1, agg, N);
  scatter_kernel<<<scatGrid, TPB, 0, stream>>>(hbuf, src, dst, enorm, agg, E);

  // layer 2: K=256, relu fused into GEMM input read
  gemm_wmma_kernel<256, true><<<gemmGrid, TPB, 0, stream>>>(agg, W2, hbuf, N);
  init_agg_kernel<<<featGrid, TPB, 0, stream>>>(hbuf, dinv, b2, agg, N);
  scatter_kernel<<<scatGrid, TPB, 0, stream>>>(hbuf, src, dst, enorm, agg, E);

  // layer 3: K=256, relu fused, aggregate directly into d_out (no final relu)
  gemm_wmma_kernel<256, true><<<gemmGrid, TPB, 0, stream>>>(agg, W3, hbuf, N);
  init_agg_kernel<<<featGrid, TPB, 0, stream>>>(hbuf, dinv, b3, out, N);
  scatter_kernel<<<scatGrid, TPB, 0, stream>>>(hbuf, src, dst, enorm, out, E);
}